// SymmetricGrpEquivLinear_14173392077445
// MI455X (gfx1250) — compile-verified
//
#include <hip/hip_runtime.h>
#include <hip/hip_bf16.h>

typedef float v2f __attribute__((ext_vector_type(2)));
typedef float v8f __attribute__((ext_vector_type(8)));

// One workgroup per batch row k. X row viewed as 64x64 matrix X(c,d).
// out(a,b) assembled from partition-structured marginals; B input is never read.
__global__ __launch_bounds__(256) void symm_equiv_kernel(
    const float* __restrict__ X,      // (1024, 4096)
    const float* __restrict__ wts,    // (15,)
    float* __restrict__ out)          // (1024, 4096)
{
    const int k    = blockIdx.x;
    const int tid  = threadIdx.x;
    const int lane = tid & 31;
    const int wave = tid >> 5;

    __shared__ __align__(16) float Xs[4096];
    __shared__ float Rc[64];   // row sums   Rc[c] = sum_d X(c,d)
    __shared__ float Cd[64];   // col sums   Cd[d] = sum_c X(c,d)
    __shared__ float Dg[64];   // diagonal   Dg[e] = X(e,e)
    __shared__ float Wl[16];
    __shared__ float Sc[2];    // {S, TD}

    // ---- load X row (16 KB) coalesced as float4 ----
    const float4* Xv = (const float4*)(X + (size_t)k * 4096);
    float4* Xsv = (float4*)Xs;
    #pragma unroll
    for (int i = 0; i < 4; ++i)
        Xsv[tid + 256 * i] = Xv[tid + 256 * i];
    if (tid < 15) Wl[tid] = wts[tid];
    __syncthreads();

    // ---- Rc / Cd via v_wmma_f32_16x16x4_f32 with B = ones (layout-invariant) ----
    // waves 0..3: row sums of X  (tile t = wave)
    // waves 4..7: row sums of X^T = col sums of X (tile t = wave-4)
    // Branchless addressing: base/stride selected per lane, no control flow in loop.
    const int  t     = wave & 3;
    const bool doCol = (wave >= 4);
    const int  mrow  = 16 * t + (lane & 15);     // M index of this lane's A rows
    const int  khalf = (lane >> 4) << 1;         // lanes 16-31 hold K+2,K+3

    const int kstr = doCol ? 64 : 1;             // element stride along K
    const int step = kstr << 2;                  // advance 4 K per WMMA
    int addr = (doCol ? mrow : mrow * 64) + khalf * kstr;

    v2f onesv; onesv[0] = 1.0f; onesv[1] = 1.0f;
    v8f acc = {0.f, 0.f, 0.f, 0.f, 0.f, 0.f, 0.f, 0.f};

    v2f a, an;
    a[0] = Xs[addr];
    a[1] = Xs[addr + kstr];
    #pragma unroll
    for (int s = 0; s < 16; ++s) {
        if (s < 15) {                            // prefetch next A while WMMA runs
            addr += step;
            an[0] = Xs[addr];
            an[1] = Xs[addr + kstr];
        }
        acc = __builtin_amdgcn_wmma_f32_16x16x4_f32(
            /*neg_a=*/false, a, /*neg_b=*/false, onesv,
            /*c_mod=*/(short)0, acc, /*reuse_a=*/false, /*reuse_b=*/false);
        a = an;
    }

    // D layout: VGPR r, lanes 0-15 -> (M=r, N=lane); lanes 16-31 -> (M=8+r).
    // Columns are identical (B=ones), so lane 0 / lane 16 carry the sums.
    {
        float* dst = doCol ? Cd : Rc;
        if (lane == 0) {
            #pragma unroll
            for (int r = 0; r < 8; ++r) dst[16 * t + r] = acc[r];
        } else if (lane == 16) {
            #pragma unroll
            for (int r = 0; r < 8; ++r) dst[16 * t + 8 + r] = acc[r];
        }
    }
    if (tid < 64) Dg[tid] = Xs[tid * 65];
    __syncthreads();

    // ---- scalars S = sum(Rc), TD = sum(Dg) via wave reduce ----
    if (wave == 0) {
        float s = Rc[lane] + Rc[lane + 32];
        #pragma unroll
        for (int off = 16; off > 0; off >>= 1) s += __shfl_down(s, off, 32);
        if (lane == 0) Sc[0] = s;
    } else if (wave == 1) {
        float s = Dg[lane] + Dg[lane + 32];
        #pragma unroll
        for (int off = 16; off > 0; off >>= 1) s += __shfl_down(s, off, 32);
        if (lane == 0) Sc[1] = s;
    }
    __syncthreads();

    // ---- assemble out(a,b); weight->partition mapping from _set_partitions trace ----
    const float S  = Sc[0];
    const float TD = Sc[1];
    const float base = Wl[14] * S + Wl[4] * TD;
    float* O = out + (size_t)k * 4096;

    #pragma unroll
    for (int i = 0; i < 16; ++i) {
        const int j = tid + 256 * i;
        const int a2 = j >> 6;
        const int b2 = j & 63;
        float v = base
                + Wl[12] * Rc[a2] + Wl[13] * Cd[a2] + Wl[3] * Dg[a2]
                + Wl[7]  * Rc[b2] + Wl[10] * Cd[b2] + Wl[1] * Dg[b2]
                + Wl[8]  * Xs[j]  + Wl[6]  * Xs[b2 * 64 + a2];
        if (a2 == b2)
            v += Wl[11] * S + Wl[2] * TD + Wl[5] * Rc[a2] + Wl[9] * Cd[a2] + Wl[0] * Dg[a2];
        O[j] = v;
    }
}

extern "C" void kernel_launch(void* const* d_in, const int* in_sizes, int n_in,
                              void* d_out, int out_size, void* d_ws, size_t ws_size,
                              hipStream_t stream) {
    const float* X   = (const float*)d_in[0];   // (1024, 4096) f32
    const float* wts = (const float*)d_in[1];   // (15,) f32
    // d_in[2] = B (15 x 4096 x 4096) is intentionally unused: its structure is
    // reproduced analytically from the partition lattice.
    float* out = (float*)d_out;                 // (1024, 4096) f32

    const int BATCH = in_sizes[0] / 4096;       // 1024
    symm_equiv_kernel<<<BATCH, 256, 0, stream>>>(X, wts, out);
}